// CWAUCHLoss_11089605558400
// MI455X (gfx1250) — compile-verified
//
#include <hip/hip_runtime.h>

// CWAUC-H loss for MI455X (gfx1250, wave32).
//
// Reference is O(B^2) pairwise; with LAMB=2 it factors exactly into 8 scalar
// reductions over B=8192 elements (~96KB input) -> single-workgroup,
// single-launch, deterministic kernel. Cross-wave reduction is done with
// V_WMMA_F32_16X16X4_F32 (exact f32 matrix unit) using an all-ones B matrix
// as a row-sum engine (ones makes the B VGPR layout irrelevant).

typedef __attribute__((ext_vector_type(2))) float v2f;
typedef __attribute__((ext_vector_type(8))) float v8f;

#define NQ    8     // number of scalar reductions
#define WPAD  9     // LDS row pad for the [32 waves][8 quantities] array

__global__ __launch_bounds__(1024)
void cwauc_loss_kernel(const float* __restrict__ output,   // (B,2) row-major
                       const float* __restrict__ labels,   // (B,1)
                       float* __restrict__ out,            // [cls, penalty]
                       int B)
{
    __shared__ float W[32 * WPAD];   // per-wave partials W[wave][q]
    __shared__ float Dld[32 * 8];    // D-matrix dump from wave 0

    const int tid  = threadIdx.x;
    const int lane = tid & 31;
    const int wave = tid >> 5;

    // ---------------- Stage 0: per-thread accumulation ----------------
    float acc[NQ];
#pragma unroll
    for (int q = 0; q < NQ; ++q) acc[q] = 0.0f;

    for (int i = tid; i < B; i += 1024) {
        const float o   = output[2 * i];      // column 0 of (B,2)
        const float lab = labels[i];
        const float pos = (lab == 1.0f) ? 1.0f : 0.0f;
        const float neg = (lab == 0.0f) ? 1.0f : 0.0f;

        const float s  = 1.0f / (1.0f + expf(-o));   // sigmoid(output)
        const float t  = 1.0f - s;
        const float s2 = 1.0f / (1.0f + expf(-s));   // sigmoid(sigmoid(output))

        acc[0] += pos;                               // Np
        acc[1] += neg;                               // Nn
        acc[2] += neg * s;                           // Sn   = sum_neg s
        acc[3] += neg * s * s;                       // Sn2  = sum_neg s^2
        acc[4] += pos * t;                           // Tp   = sum_pos (1-s)
        acc[5] += pos * t * t;                       // Tp2  = sum_pos (1-s)^2
        acc[6] += pos * logf(s2 + 1e-32f);           // PL
        acc[7] += neg * logf(1.0f - s2 + 1e-32f);    // NL
    }

    // ---------------- Stage 1: intra-wave reduction (wave32 shuffles) ----
#pragma unroll
    for (int q = 0; q < NQ; ++q) {
        float v = acc[q];
#pragma unroll
        for (int off = 16; off >= 1; off >>= 1)
            v += __shfl_xor(v, off, 32);
        acc[q] = v;
    }
    if (lane == 0) {
#pragma unroll
        for (int q = 0; q < NQ; ++q) W[wave * WPAD + q] = acc[q];
    }
    __syncthreads();

    // ---------------- Stage 2: cross-wave reduction via f32 WMMA ---------
    // A_i (16x4 f32): A_i[m][k] = W[8*i + 4*r + k][q], where m = 2*q + r.
    // B = all ones (4x16) => D[m][n] = sum_k A_i[m][k], accumulated over i.
    // Every wave executes this (EXEC all-1s as WMMA requires).
    //
    // ISA A-layout (32-bit A 16x4): lanes 0-15 hold K=0(V0),1(V1) of row M=lane;
    // lanes 16-31 hold K=2(V0),3(V1) of row M=lane-16.
    const int m     = lane & 15;
    const int kbase = (lane >> 4) * 2;   // 0 or 2
    const int q     = m >> 1;
    const int r     = m & 1;
    const int wbase = 4 * r + kbase;

    v8f d = {};
    v2f bones;
    bones.x = 1.0f;
    bones.y = 1.0f;
#pragma unroll
    for (int i = 0; i < 4; ++i) {
        v2f a;
        a.x = W[(8 * i + wbase)     * WPAD + q];   // A[m][kbase]
        a.y = W[(8 * i + wbase + 1) * WPAD + q];   // A[m][kbase+1]
        // (neg_a, A, neg_b, B, c_mod, C, reuse_a, reuse_b)
        d = __builtin_amdgcn_wmma_f32_16x16x4_f32(
                false, a, false, bones, (short)0, d, false, false);
    }

    if (wave == 0) {
#pragma unroll
        for (int j = 0; j < 8; ++j) Dld[lane * 8 + j] = d[j];
    }
    __syncthreads();

    // ---------------- Stage 3: final scalars (thread 0) ------------------
    if (tid == 0) {
        // D VGPR layout: VGPR j @ lanes 0-15 = row j; @ lanes 16-31 = row j+8.
        // sum[q] = D[2q][*] + D[2q+1][*]
        float sum[NQ];
#pragma unroll
        for (int qq = 0; qq < NQ; ++qq) {
            if (qq < 4)
                sum[qq] = Dld[0 * 8 + 2 * qq] + Dld[0 * 8 + 2 * qq + 1];
            else
                sum[qq] = Dld[16 * 8 + (2 * qq - 8)] + Dld[16 * 8 + (2 * qq - 7)];
        }
        const float Np = sum[0], Nn = sum[1];
        const float Sn = sum[2], Sn2 = sum[3];
        const float Tp = sum[4], Tp2 = sum[5];
        const float PL = sum[6], NL = sum[7];

        // sum_{i neg, j pos} (1 - s_j + s_i)^2, factored (exact):
        const float pairSum = Np * Sn2 + 2.0f * Sn * Tp + Nn * Tp2;
        const float penalty = pairSum / (Np * Nn) / 2.0f;   // / LAMB

        const float Bf     = (float)B;
        const float beta_p = Np / (Np + Nn);
        const float beta_n = Nn / (Np + Nn);
        const float fpcls  = -beta_n * (PL / Bf) - beta_p * (NL / Bf);

        out[0] = fpcls + penalty;   // cls  (ALPHA = 1)
        out[1] = penalty;           // ALPHA * penalty
    }
}

extern "C" void kernel_launch(void* const* d_in, const int* in_sizes, int n_in,
                              void* d_out, int out_size, void* d_ws, size_t ws_size,
                              hipStream_t stream) {
    const float* output = (const float*)d_in[0];
    const float* labels = (const float*)d_in[1];
    float* out = (float*)d_out;
    const int B = in_sizes[1];   // labels element count = 8192

    hipLaunchKernelGGL(cwauc_loss_kernel, dim3(1), dim3(1024), 0, stream,
                       output, labels, out, B);
}